// SS2D_12249246728874
// MI455X (gfx1250) — compile-verified
//
#include <hip/hip_runtime.h>
#include <hip/hip_bf16.h>

// ---------------------------------------------------------------------------
// SS2D (Mamba selective-scan 2D) block for gfx1250 (MI455X).
//   1. wmma_gemm<2,4> : xz   = x @ in_proj_w^T + b        (WMMA f32 16x16x4)
//   2. dwconv_silu    : u    = silu(dwconv3x3(xin) + cb)
//   3. wmma_gemm<2,3> : xdbl = u @ x_proj_w^T             (dt|B|C, padded 48)
//   4. wmma_gemm<2,4> : delta= softplus(dt @ dt_proj_w^T + dt_b)
//   5. chunked selective scan (3 passes, 32 chunks of 128 -> 12288 waves)
//   6. ln_gate        : g = LN(y)*silu(z)
//   7. wmma_gemm<2,4> : out = g @ out_proj_w^T + b
// ---------------------------------------------------------------------------

#define D_MODEL 192
#define D_INNER 384
#define D_STATE 16
#define DT_RANK 12
#define NB      2
#define HH      64
#define WW      64
#define LSEQ    (HH * WW)          // 4096
#define MROWS   (NB * LSEQ)        // 8192
#define XDBL_LD 48                 // 44 padded to 48 (dt:0-11, B:12-27, C:28-43)
#define NCHAIN  (NB * D_INNER)     // 768
#define CHUNKS  32
#define CLEN    (LSEQ / CHUNKS)    // 128

typedef __attribute__((ext_vector_type(2))) float v2f;
typedef __attribute__((ext_vector_type(8))) float v8f;

__device__ __forceinline__ float softplus_f(float x) {
    return x > 20.f ? x : log1pf(__expf(x));
}
__device__ __forceinline__ float silu_f(float x) {
    return x / (1.f + __expf(-x));
}

// ---------------------------------------------------------------------------
// C[M,N] = A[M,K] * W[N,K]^T (+bias) (+softplus).  V_WMMA_F32_16X16X4_F32.
// Register-blocked: each wave computes MT x NT 16x16 tiles (A frags reused
// across NT, B frags reused across MT -> 8 WMMAs per 6 b64 loads at 2x4).
// Padded B-columns are only clamped (never masked): a garbage B column only
// affects output columns >= N, whose stores are guarded.
// ---------------------------------------------------------------------------
template <int MT, int NT, int ACT, bool NG>
__global__ void wmma_gemm(const float* __restrict__ A,
                          const float* __restrict__ W,
                          const float* __restrict__ bias,
                          float* __restrict__ C,
                          int M, int N, int K,
                          int lda, int ldw, int ldc) {
    const int lane = threadIdx.x & 31;
    const int wv   = threadIdx.x >> 5;
    const int ntg  = ((N + 15) >> 4) / NT;   // n-tile groups
    const int mtg  = (M >> 4) / MT;          // m-tile groups
    const int grp  = blockIdx.x * (blockDim.x >> 5) + wv;
    if (grp >= ntg * mtg) return;            // wave-uniform: EXEC stays all-1s
    const int ng = grp % ntg;
    const int mg = grp / ntg;

    const int r16   = lane & 15;
    const int half  = lane >> 4;
    const int khalf = half << 1;             // A/B frag K offset: 0 or 2

    const float* ap[MT];
    const float* wp[NT];
#pragma unroll
    for (int i = 0; i < MT; ++i) {
        const int mrow = ((mg * MT + i) << 4) + r16;
        ap[i] = A + (size_t)mrow * lda + khalf;
    }
#pragma unroll
    for (int j = 0; j < NT; ++j) {
        int ncol = ((ng * NT + j) << 4) + r16;
        if (NG && ncol >= N) ncol = N - 1;   // clamp only; no mask needed
        wp[j] = W + (size_t)ncol * ldw + khalf;
    }

    v8f acc[MT][NT];
#pragma unroll
    for (int i = 0; i < MT; ++i)
#pragma unroll
        for (int j = 0; j < NT; ++j) acc[i][j] = (v8f){};

    for (int k = 0; k < K; k += 4) {
        v2f av[MT], bv[NT];
#pragma unroll
        for (int i = 0; i < MT; ++i) av[i] = *(const v2f*)(ap[i] + k);
#pragma unroll
        for (int j = 0; j < NT; ++j) bv[j] = *(const v2f*)(wp[j] + k);
#pragma unroll
        for (int i = 0; i < MT; ++i)
#pragma unroll
            for (int j = 0; j < NT; ++j)
                acc[i][j] = __builtin_amdgcn_wmma_f32_16x16x4_f32(
                    false, av[i], false, bv[j], (short)0, acc[i][j],
                    false, false);
    }

#pragma unroll
    for (int i = 0; i < MT; ++i) {
        const int rbase = ((mg * MT + i) << 4) + (half << 3);
#pragma unroll
        for (int j = 0; j < NT; ++j) {
            const int col = ((ng * NT + j) << 4) + r16;
            if (!NG || col < N) {
                float b = bias ? bias[col] : 0.f;
#pragma unroll
                for (int r = 0; r < 8; ++r) {
                    float v = acc[i][j][r] + b;
                    if (ACT == 1) v = softplus_f(v);
                    C[(size_t)(rbase + r) * ldc + col] = v;
                }
            }
        }
    }
}

// ---------------------------------------------------------------------------
// Depthwise 3x3 conv (zero pad) + bias + SiLU.  xin = xz[:, 0:384].
// ---------------------------------------------------------------------------
__global__ void dwconv_silu_kernel(const float* __restrict__ xz,
                                   const float* __restrict__ cw,   // [384,1,3,3]
                                   const float* __restrict__ cb,
                                   float* __restrict__ u) {
    const int d = threadIdx.x;       // 0..383
    const int m = blockIdx.x;        // 0..8191
    const int w = m & (WW - 1);
    const int h = (m >> 6) & (HH - 1);
    const int b = m >> 12;
    const float* wp = cw + d * 9;
    float acc = cb[d];
#pragma unroll
    for (int dy = 0; dy < 3; ++dy) {
        const int hh = h + dy - 1;
        if ((unsigned)hh >= (unsigned)HH) continue;
#pragma unroll
        for (int dx = 0; dx < 3; ++dx) {
            const int ww2 = w + dx - 1;
            if ((unsigned)ww2 >= (unsigned)WW) continue;
            const int mm = (b << 12) + (hh << 6) + ww2;
            acc = fmaf(xz[(size_t)mm * (2 * D_INNER) + d], wp[dy * 3 + dx], acc);
        }
    }
    u[(size_t)m * D_INNER + d] = silu_f(acc);
}

// ---------------------------------------------------------------------------
// Chunked selective scan.  One 16-lane half-wave owns one (chain, chunk);
// lane n holds state n.  h_l = exp(delta_l*A_n)*h_{l-1} + delta_l*u_l*B_ln.
// Pass 1: local scan (h0=0), record final state hfin and decay product P.
// Pass 2: per chain, sequential combine of 32 chunk summaries -> hinit.
// Pass 3: re-scan each chunk from hinit, emit y = <h,C> + u*D.
// ---------------------------------------------------------------------------
__global__ void scan_pass1(const float* __restrict__ delta,
                           const float* __restrict__ u,
                           const float* __restrict__ xdbl,
                           const float* __restrict__ A_log,
                           float* __restrict__ hfin,
                           float* __restrict__ Pout) {
    const int lane = threadIdx.x & 31;
    const int wv   = threadIdx.x >> 5;
    const int n    = lane & 15;
    const int half = lane >> 4;
    const int base = half << 4;
    const int slot = (blockIdx.x * (blockDim.x >> 5) + wv) * 2 + half; // 0..24575
    const int chunk = slot & (CHUNKS - 1);
    const int chain = slot >> 5;
    const int b = chain / D_INNER;
    const int d = chain - b * D_INNER;

    const float a_coef = -__expf(A_log[d * D_STATE + n]);
    float h = 0.f, P = 1.f;
    const int l0 = chunk * CLEN;
    for (int l = l0; l < l0 + CLEN; ++l) {
        const int m = b * LSEQ + l;
        float t = 0.f;
        if (n == 0) t = delta[(size_t)m * D_INNER + d];
        if (n == 1) t = u[(size_t)m * D_INNER + d];
        const float dl = __shfl(t, base + 0, 32);
        const float ul = __shfl(t, base + 1, 32);
        const float Bn = xdbl[(size_t)m * XDBL_LD + DT_RANK + n];
        const float dA = __expf(dl * a_coef);
        h = fmaf(dA, h, dl * ul * Bn);
        P *= dA;
    }
    hfin[(size_t)slot * D_STATE + n] = h;
    Pout[(size_t)slot * D_STATE + n] = P;
}

__global__ void scan_combine(const float* __restrict__ hfin,
                             const float* __restrict__ P,
                             float* __restrict__ hinit) {
    const int lane = threadIdx.x & 31;
    const int wv   = threadIdx.x >> 5;
    const int n    = lane & 15;
    const int half = lane >> 4;
    const int chain = (blockIdx.x * (blockDim.x >> 5) + wv) * 2 + half; // 0..767
    float run = 0.f;
    for (int j = 0; j < CHUNKS; ++j) {
        const size_t idx = ((size_t)chain * CHUNKS + j) * D_STATE + n;
        hinit[idx] = run;
        run = fmaf(P[idx], run, hfin[idx]);
    }
}

__global__ void scan_pass3(const float* __restrict__ delta,
                           const float* __restrict__ u,
                           const float* __restrict__ xdbl,
                           const float* __restrict__ A_log,
                           const float* __restrict__ Ds,
                           const float* __restrict__ hinit,
                           float* __restrict__ y) {
    const int lane = threadIdx.x & 31;
    const int wv   = threadIdx.x >> 5;
    const int n    = lane & 15;
    const int half = lane >> 4;
    const int base = half << 4;
    const int slot = (blockIdx.x * (blockDim.x >> 5) + wv) * 2 + half;
    const int chunk = slot & (CHUNKS - 1);
    const int chain = slot >> 5;
    const int b = chain / D_INNER;
    const int d = chain - b * D_INNER;

    const float a_coef = -__expf(A_log[d * D_STATE + n]);
    const float dcoef  = Ds[d];
    float h = hinit[(size_t)slot * D_STATE + n];
    const int l0 = chunk * CLEN;
    for (int l = l0; l < l0 + CLEN; ++l) {
        const int m = b * LSEQ + l;
        float t = 0.f;
        if (n == 0) t = delta[(size_t)m * D_INNER + d];
        if (n == 1) t = u[(size_t)m * D_INNER + d];
        const float dl = __shfl(t, base + 0, 32);
        const float ul = __shfl(t, base + 1, 32);
        const float Bn = xdbl[(size_t)m * XDBL_LD + DT_RANK + n];
        const float Cn = xdbl[(size_t)m * XDBL_LD + DT_RANK + D_STATE + n];
        h = fmaf(__expf(dl * a_coef), h, dl * ul * Bn);
        float p = h * Cn;
        p += __shfl_xor(p, 1, 32);
        p += __shfl_xor(p, 2, 32);
        p += __shfl_xor(p, 4, 32);
        p += __shfl_xor(p, 8, 32);
        if (n == 0) y[(size_t)m * D_INNER + d] = fmaf(ul, dcoef, p);
    }
}

// ---------------------------------------------------------------------------
// LayerNorm(384) * gamma + beta, then * silu(z).  One block per row (12 waves).
// ---------------------------------------------------------------------------
__global__ void ln_gate_kernel(const float* __restrict__ y,
                               const float* __restrict__ xz,
                               const float* __restrict__ ln_g,
                               const float* __restrict__ ln_b,
                               float* __restrict__ g) {
    __shared__ float ss[12];
    __shared__ float ssq[12];
    const int m = blockIdx.x;
    const int d = threadIdx.x;         // 0..383
    const float v = y[(size_t)m * D_INNER + d];
    float s = v, sq = v * v;
#pragma unroll
    for (int off = 1; off < 32; off <<= 1) {
        s  += __shfl_xor(s, off, 32);
        sq += __shfl_xor(sq, off, 32);
    }
    const int wv = threadIdx.x >> 5;
    if ((threadIdx.x & 31) == 0) { ss[wv] = s; ssq[wv] = sq; }
    __syncthreads();
    float S = 0.f, SQ = 0.f;
#pragma unroll
    for (int i = 0; i < 12; ++i) { S += ss[i]; SQ += ssq[i]; }
    const float mu  = S * (1.f / D_INNER);
    const float var = SQ * (1.f / D_INNER) - mu * mu;
    const float nv  = (v - mu) * rsqrtf(var + 1e-5f) * ln_g[d] + ln_b[d];
    const float zz  = xz[(size_t)m * (2 * D_INNER) + D_INNER + d];
    g[(size_t)m * D_INNER + d] = nv * silu_f(zz);
}

// ---------------------------------------------------------------------------
extern "C" void kernel_launch(void* const* d_in, const int* in_sizes, int n_in,
                              void* d_out, int out_size, void* d_ws, size_t ws_size,
                              hipStream_t stream) {
    const float* x          = (const float*)d_in[0];
    const float* in_proj_w  = (const float*)d_in[1];
    const float* in_proj_b  = (const float*)d_in[2];
    const float* conv_w     = (const float*)d_in[3];
    const float* conv_b     = (const float*)d_in[4];
    const float* x_proj_w   = (const float*)d_in[5];
    const float* dt_proj_w  = (const float*)d_in[6];
    const float* dt_proj_b  = (const float*)d_in[7];
    const float* A_log      = (const float*)d_in[8];
    const float* Ds         = (const float*)d_in[9];
    const float* ln_g       = (const float*)d_in[10];
    const float* ln_b       = (const float*)d_in[11];
    const float* out_proj_w = (const float*)d_in[12];
    const float* out_proj_b = (const float*)d_in[13];
    float* out = (float*)d_out;

    float* ws    = (float*)d_ws;
    float* xz    = ws;                                  // 8192*768
    float* u     = xz + (size_t)MROWS * 2 * D_INNER;    // 8192*384
    float* xdbl  = u + (size_t)MROWS * D_INNER;         // 8192*48
    float* delta = xdbl + (size_t)MROWS * XDBL_LD;      // 8192*384
    float* yb    = delta + (size_t)MROWS * D_INNER;     // 8192*384
    float* hfin  = yb + (size_t)MROWS * D_INNER;        // 768*32*16
    float* Pbuf  = hfin + (size_t)NCHAIN * CHUNKS * D_STATE;
    float* hinit = Pbuf + (size_t)NCHAIN * CHUNKS * D_STATE;
    float* g     = delta;                               // reuse delta after scan

    const int WPB = 4, TPB = WPB * 32;
    auto blocks = [&](int M, int N, int MT, int NT) {
        int grp = ((M >> 4) / MT) * ((((N + 15) >> 4)) / NT);
        return (grp + WPB - 1) / WPB;
    };

    // 1. in_proj: xz = x @ W^T + b     [8192,768] = [8192,192] x [768,192]^T
    wmma_gemm<2, 4, 0, false><<<blocks(MROWS, 2 * D_INNER, 2, 4), TPB, 0, stream>>>(
        x, in_proj_w, in_proj_b, xz,
        MROWS, 2 * D_INNER, D_MODEL, D_MODEL, D_MODEL, 2 * D_INNER);

    // 2. depthwise conv + SiLU
    dwconv_silu_kernel<<<MROWS, D_INNER, 0, stream>>>(xz, conv_w, conv_b, u);

    // 3. x_proj: xdbl = u @ W^T        [8192,44] = [8192,384] x [44,384]^T
    wmma_gemm<2, 3, 0, true><<<blocks(MROWS, DT_RANK + 2 * D_STATE, 2, 3), TPB, 0, stream>>>(
        u, x_proj_w, nullptr, xdbl,
        MROWS, DT_RANK + 2 * D_STATE, D_INNER, D_INNER, D_INNER, XDBL_LD);

    // 4. dt_proj + softplus: delta = softplus(dt @ W^T + b)
    wmma_gemm<2, 4, 1, false><<<blocks(MROWS, D_INNER, 2, 4), TPB, 0, stream>>>(
        xdbl, dt_proj_w, dt_proj_b, delta,
        MROWS, D_INNER, DT_RANK, XDBL_LD, DT_RANK, D_INNER);

    // 5. chunked selective scan: 768 chains x 32 chunks
    scan_pass1<<<(NCHAIN * CHUNKS) / 8, 128, 0, stream>>>(delta, u, xdbl, A_log,
                                                          hfin, Pbuf);
    scan_combine<<<NCHAIN / 8, 128, 0, stream>>>(hfin, Pbuf, hinit);
    scan_pass3<<<(NCHAIN * CHUNKS) / 8, 128, 0, stream>>>(delta, u, xdbl, A_log, Ds,
                                                          hinit, yb);

    // 6. LayerNorm + gate
    ln_gate_kernel<<<MROWS, D_INNER, 0, stream>>>(yb, xz, ln_g, ln_b, g);

    // 7. out_proj: out = g @ W^T + b   [8192,192] = [8192,384] x [192,384]^T
    wmma_gemm<2, 4, 0, false><<<blocks(MROWS, D_MODEL, 2, 4), TPB, 0, stream>>>(
        g, out_proj_w, out_proj_b, out,
        MROWS, D_MODEL, D_INNER, D_INNER, D_INNER, D_MODEL);
}